// UNetV2_88210038325723
// MI455X (gfx1250) — compile-verified
//
#include <hip/hip_runtime.h>

// -------------------------------------------------------------------------
// Submanifold sparse-conv stack on MI455X (gfx1250, wave32).
// One wave <-> one tile of 16 voxels x 16 output channels, accumulated with
// V_WMMA_F32_16X16X4_F32 (exact fp32 matrix op, matches the fp32 reference).
//
// Gathers are branchless: index clamped to 0, unconditional float2 load,
// fragment multiplied by the 0/1 mask — exactly the reference's
// take(feats, max(nbr,0)) * mask semantics, with no EXEC manipulation so the
// load stream pipelines and the WMMAs don't serialize behind branches.
// Two accumulators (even/odd k) break the single D->C WMMA chain.
// -------------------------------------------------------------------------

typedef float v2f __attribute__((ext_vector_type(2)));
typedef float v8f __attribute__((ext_vector_type(8)));

#define N_VOXELS 200000
#define NTILES   ((N_VOXELS + 15) / 16)      // 12500 (exact)
#define NK       27
#define WAVES_PB 8
#define BLOCK    (WAVES_PB * 32)

// ---------------- Layer 0: C_in = 4, scale/bias/ReLU ----------------------
__global__ __launch_bounds__(BLOCK) void subm_conv_c4(
    const float* __restrict__ feats,   // [N,4]
    const int*   __restrict__ nbr,     // [N,27]
    const float* __restrict__ W,       // [27,4,16]
    const float* __restrict__ s,       // [16]
    const float* __restrict__ b,       // [16]
    float*       __restrict__ out)     // [N,16]
{
    __shared__ float sW[NK * 4 * 16];  // 6.75 KB
    for (int i = threadIdx.x; i < NK * 4 * 16; i += BLOCK) sW[i] = W[i];
    __syncthreads();

    const int lane = threadIdx.x & 31;
    const int wave = threadIdx.x >> 5;
    const int m    = lane & 15;        // col / voxel-row within half-wave
    const int hi   = lane >> 4;        // half-wave select (K split)
    const int t    = blockIdx.x * WAVES_PB + wave;
    if (t >= NTILES) return;

    const int* nrow = nbr + (size_t)(t * 16 + m) * NK;
    v8f acc0 = {};
    v8f acc1 = {};

    #pragma unroll
    for (int k = 0; k < NK; ++k) {
        const int   idx = nrow[k];
        const float msk = (idx >= 0) ? 1.0f : 0.0f;
        const int   cid = (idx >= 0) ? idx : 0;
        v2f a = *(const v2f*)(feats + (size_t)cid * 4 + 2 * hi); // K=(2hi,2hi+1)
        a.x *= msk;
        a.y *= msk;
        const float* wk = sW + (k * 4 + 2 * hi) * 16 + m;
        v2f bb = { wk[0], wk[16] };
        if (k & 1)
            acc1 = __builtin_amdgcn_wmma_f32_16x16x4_f32(
                       false, a, false, bb, (short)0, acc1, false, false);
        else
            acc0 = __builtin_amdgcn_wmma_f32_16x16x4_f32(
                       false, a, false, bb, (short)0, acc0, false, false);
    }
    const v8f acc = acc0 + acc1;

    const float sc = s[m], bi = b[m];
    #pragma unroll
    for (int j = 0; j < 8; ++j) {
        const int r = t * 16 + j + 8 * hi;
        float v = acc[j] * sc + bi;
        v = v > 0.0f ? v : 0.0f;
        out[(size_t)r * 16 + m] = v;
    }
}

// ---------------- Layer 1: C_in = 16, scale/bias/ReLU ---------------------
__global__ __launch_bounds__(BLOCK) void subm_conv_c16_relu(
    const float* __restrict__ feats,   // [N,16]
    const int*   __restrict__ nbr,     // [N,27]
    const float* __restrict__ W,       // [27,16,16]
    const float* __restrict__ s,
    const float* __restrict__ b,
    float*       __restrict__ out)     // [N,16]
{
    __shared__ float sW[NK * 16 * 16]; // 27 KB of the 320 KB WGP LDS
    for (int i = threadIdx.x; i < NK * 16 * 16; i += BLOCK) sW[i] = W[i];
    __syncthreads();

    const int lane = threadIdx.x & 31;
    const int wave = threadIdx.x >> 5;
    const int m    = lane & 15;
    const int hi   = lane >> 4;
    const int t    = blockIdx.x * WAVES_PB + wave;
    if (t >= NTILES) return;

    const int* nrow = nbr + (size_t)(t * 16 + m) * NK;
    v8f acc0 = {};
    v8f acc1 = {};

    for (int k = 0; k < NK; ++k) {
        const int   idx = nrow[k];
        const float msk = (idx >= 0) ? 1.0f : 0.0f;
        const int   cid = (idx >= 0) ? idx : 0;
        const float* g  = feats + (size_t)cid * 16 + 2 * hi;

        // batch all four K-chunk gathers (8B each), then mask
        v2f a0 = *(const v2f*)(g + 0);
        v2f a1 = *(const v2f*)(g + 4);
        v2f a2 = *(const v2f*)(g + 8);
        v2f a3 = *(const v2f*)(g + 12);
        a0.x *= msk; a0.y *= msk;
        a1.x *= msk; a1.y *= msk;
        a2.x *= msk; a2.y *= msk;
        a3.x *= msk; a3.y *= msk;

        const float* wk = sW + (k * 16 + 2 * hi) * 16 + m;
        v2f b0 = { wk[0 * 16],  wk[1 * 16]  + 0.0f * wk[0],  };
        b0.x = wk[0];        b0.y = wk[16];
        v2f b1 = { wk[64],   wk[64 + 16] };
        v2f b2 = { wk[128],  wk[128 + 16] };
        v2f b3 = { wk[192],  wk[192 + 16] };

        acc0 = __builtin_amdgcn_wmma_f32_16x16x4_f32(
                   false, a0, false, b0, (short)0, acc0, false, false);
        acc1 = __builtin_amdgcn_wmma_f32_16x16x4_f32(
                   false, a1, false, b1, (short)0, acc1, false, false);
        acc0 = __builtin_amdgcn_wmma_f32_16x16x4_f32(
                   false, a2, false, b2, (short)0, acc0, false, false);
        acc1 = __builtin_amdgcn_wmma_f32_16x16x4_f32(
                   false, a3, false, b3, (short)0, acc1, false, false);
    }
    const v8f acc = acc0 + acc1;

    const float sc = s[m], bi = b[m];
    #pragma unroll
    for (int j = 0; j < 8; ++j) {
        const int r = t * 16 + j + 8 * hi;
        float v = acc[j] * sc + bi;
        v = v > 0.0f ? v : 0.0f;
        out[(size_t)r * 16 + m] = v;
    }
}

// ------- Layer 2: C_in = 16, scale/bias + residual + ReLU + 16->3 linear --
__global__ __launch_bounds__(BLOCK) void subm_conv_c16_res_linear(
    const float* __restrict__ feats,   // h : [N,16]
    const int*   __restrict__ nbr,     // [N,27]
    const float* __restrict__ W,       // [27,16,16]
    const float* __restrict__ s,
    const float* __restrict__ b,
    const float* __restrict__ resid,   // x0 : [N,16]
    const float* __restrict__ Wlin,    // [16,3]
    const float* __restrict__ blin,    // [3]
    float*       __restrict__ out)     // [N,3]
{
    __shared__ float sW[NK * 16 * 16];          // 27 KB
    __shared__ float sX[WAVES_PB][16][17];      // 8.5 KB, padded vs bank conflicts
    for (int i = threadIdx.x; i < NK * 16 * 16; i += BLOCK) sW[i] = W[i];
    __syncthreads();

    const int lane = threadIdx.x & 31;
    const int wave = threadIdx.x >> 5;
    const int m    = lane & 15;
    const int hi   = lane >> 4;
    const int t    = blockIdx.x * WAVES_PB + wave;
    const bool active = (t < NTILES);

    if (active) {
        const int* nrow = nbr + (size_t)(t * 16 + m) * NK;
        v8f acc0 = {};
        v8f acc1 = {};

        for (int k = 0; k < NK; ++k) {
            const int   idx = nrow[k];
            const float msk = (idx >= 0) ? 1.0f : 0.0f;
            const int   cid = (idx >= 0) ? idx : 0;
            const float* g  = feats + (size_t)cid * 16 + 2 * hi;

            v2f a0 = *(const v2f*)(g + 0);
            v2f a1 = *(const v2f*)(g + 4);
            v2f a2 = *(const v2f*)(g + 8);
            v2f a3 = *(const v2f*)(g + 12);
            a0.x *= msk; a0.y *= msk;
            a1.x *= msk; a1.y *= msk;
            a2.x *= msk; a2.y *= msk;
            a3.x *= msk; a3.y *= msk;

            const float* wk = sW + (k * 16 + 2 * hi) * 16 + m;
            v2f b0 = { wk[0],    wk[16] };
            v2f b1 = { wk[64],   wk[64 + 16] };
            v2f b2 = { wk[128],  wk[128 + 16] };
            v2f b3 = { wk[192],  wk[192 + 16] };

            acc0 = __builtin_amdgcn_wmma_f32_16x16x4_f32(
                       false, a0, false, b0, (short)0, acc0, false, false);
            acc1 = __builtin_amdgcn_wmma_f32_16x16x4_f32(
                       false, a1, false, b1, (short)0, acc1, false, false);
            acc0 = __builtin_amdgcn_wmma_f32_16x16x4_f32(
                       false, a2, false, b2, (short)0, acc0, false, false);
            acc1 = __builtin_amdgcn_wmma_f32_16x16x4_f32(
                       false, a3, false, b3, (short)0, acc1, false, false);
        }
        const v8f acc = acc0 + acc1;

        const float sc = s[m], bi = b[m];
        #pragma unroll
        for (int j = 0; j < 8; ++j) {
            const int r = t * 16 + j + 8 * hi;
            float v = acc[j] * sc + bi + resid[(size_t)r * 16 + m];
            v = v > 0.0f ? v : 0.0f;
            sX[wave][j + 8 * hi][m] = v;      // transpose-stage for the linear
        }
    }
    __syncthreads();

    if (active && hi == 0) {                  // lane m handles voxel t*16+m
        float y0 = blin[0], y1 = blin[1], y2 = blin[2];
        #pragma unroll
        for (int d = 0; d < 16; ++d) {
            const float xv = sX[wave][m][d];
            y0 += xv * Wlin[d * 3 + 0];
            y1 += xv * Wlin[d * 3 + 1];
            y2 += xv * Wlin[d * 3 + 2];
        }
        const size_t o = (size_t)(t * 16 + m) * 3;
        out[o + 0] = y0;
        out[o + 1] = y1;
        out[o + 2] = y2;
    }
}

// -------------------------------------------------------------------------
extern "C" void kernel_launch(void* const* d_in, const int* in_sizes, int n_in,
                              void* d_out, int out_size, void* d_ws, size_t ws_size,
                              hipStream_t stream) {
    const float* feats = (const float*)d_in[0];
    const int*   nbr   = (const int*)  d_in[1];
    const float* W0    = (const float*)d_in[2];
    const float* s0    = (const float*)d_in[3];
    const float* b0    = (const float*)d_in[4];
    const float* W1    = (const float*)d_in[5];
    const float* s1    = (const float*)d_in[6];
    const float* b1    = (const float*)d_in[7];
    const float* W2    = (const float*)d_in[8];
    const float* s2    = (const float*)d_in[9];
    const float* b2    = (const float*)d_in[10];
    const float* Wlin  = (const float*)d_in[11];
    const float* blin  = (const float*)d_in[12];
    float* out = (float*)d_out;

    // scratch: x0 (residual) and h, each [N,16] f32  -> 25.6 MB total
    float* x0 = (float*)d_ws;
    float* h  = x0 + (size_t)N_VOXELS * 16;

    const dim3 grid((NTILES + WAVES_PB - 1) / WAVES_PB);
    const dim3 block(BLOCK);

    subm_conv_c4<<<grid, block, 0, stream>>>(feats, nbr, W0, s0, b0, x0);
    subm_conv_c16_relu<<<grid, block, 0, stream>>>(x0, nbr, W1, s1, b1, h);
    subm_conv_c16_res_linear<<<grid, block, 0, stream>>>(h, nbr, W2, s2, b2,
                                                         x0, Wlin, blin, out);
}